// GCNModel_75548474736928
// MI455X (gfx1250) — compile-verified
//
#include <hip/hip_runtime.h>
#include <hip/hip_bf16.h>
#include <math.h>

typedef __attribute__((ext_vector_type(2))) float v2f;
typedef __attribute__((ext_vector_type(8))) float v8f;
typedef __attribute__((ext_vector_type(4))) unsigned int u32x4;
typedef __attribute__((ext_vector_type(8))) int i32x8;
typedef __attribute__((ext_vector_type(4))) int i32x4;

#define D_IN      128
#define D_HID     128
#define D_OUT     40
#define D_OUT_PAD 48

#if __has_builtin(__builtin_amdgcn_tensor_load_to_lds) && \
    __has_builtin(__builtin_amdgcn_s_wait_tensorcnt)
#define USE_TDM 1
#else
#define USE_TDM 0
#endif

// ---------------- degree / normalization ----------------

__global__ void k_init_deg(float* __restrict__ deg, int n) {
    int i = blockIdx.x * blockDim.x + threadIdx.x;
    if (i < n) deg[i] = 1.0f;                       // self-loop weight
}

__global__ void k_deg_scatter(float* __restrict__ deg, const int* __restrict__ dst,
                              const float* __restrict__ ew, int E) {
    int e = blockIdx.x * blockDim.x + threadIdx.x;
    if (e < E) unsafeAtomicAdd(&deg[dst[e]], ew[e]); // global_atomic_add_f32
}

__global__ void k_rsqrt_inplace(float* __restrict__ deg, int n) {
    int i = blockIdx.x * blockDim.x + threadIdx.x;
    if (i < n) {
        float d = deg[i];
        deg[i] = (d > 0.0f) ? rsqrtf(d) : 0.0f;
    }
}

__global__ void k_zero(float* __restrict__ p, long long n) {
    long long i = (long long)blockIdx.x * blockDim.x + threadIdx.x;
    if (i < n) p[i] = 0.0f;
}

// ---------------- WMMA f32 GEMM:  Out[N x 16CT] = X[N x 128] @ W[128 x 16CT] ----------------
// One block = 8 waves; W staged fully in LDS (via Tensor Data Mover when the
// builtin is available); each wave computes a 16-row stripe across all CT
// column tiles with V_WMMA_F32_16X16X4_F32 (K-step = 4).

template <int CT>
__global__ __launch_bounds__(256) void k_gemm_wmma(const float* __restrict__ X,
                                                   const float* __restrict__ Wm,
                                                   float* __restrict__ Out,
                                                   int rowTiles) {
    constexpr unsigned NC = 16 * CT;
    __shared__ float wlds[D_IN * NC];

#if USE_TDM
    // One wave issues a single 2-D TDM descriptor: copy W[128 x NC] f32,
    // contiguous (dim0 stride = NC), into LDS at wlds. EXEC is ignored by
    // TENSOR_LOAD_TO_LDS, so guard at wave granularity.
    if (threadIdx.x < 32) {
        unsigned long long ga = (unsigned long long)(const void*)Wm;
        unsigned lds_base = (unsigned)(size_t)(void*)wlds;       // LDS aperture: addr[31:0]
        u32x4 g0;
        g0[0] = 1u;                                              // count=1 (valid), user mode
        g0[1] = lds_base;                                        // lds_addr
        g0[2] = (unsigned)(ga & 0xFFFFFFFFu);                    // global_addr[31:0]
        g0[3] = (unsigned)((ga >> 32) & 0x01FFFFFFu) | (2u << 30); // addr[56:32] | type=2
        i32x8 g1;
        g1[0] = 0x00020000;                // data_size=2 (4B), no multicast/flags
        g1[1] = (int)((NC & 0xFFFFu) << 16);        // tensor_dim0[15:0]  @ bits 63:48
        g1[2] = (int)((unsigned)D_IN << 16);        // dim0[31:16]=0 | tensor_dim1[15:0]
        g1[3] = (int)(NC << 16);                    // dim1[31:16]=0 | tile_dim0=NC
        g1[4] = (int)D_IN;                          // tile_dim1=128 | tile_dim2=0
        g1[5] = (int)NC;                            // tensor_dim0_stride[31:0]
        g1[6] = 0;                                  // stride hi | dim1_stride lo (2-D: unused)
        g1[7] = 0;
        i32x4 z4 = {0, 0, 0, 0};                    // groups 2/3: NULL (<=2-D tensor)
        i32x8 z8 = {0, 0, 0, 0, 0, 0, 0, 0};        // extra descriptor group (unused)
        __builtin_amdgcn_tensor_load_to_lds(g0, g1, z4, z4, z8, 0);
        __builtin_amdgcn_s_wait_tensorcnt(0);       // s_wait_tensorcnt 0
    }
    __syncthreads();
#else
    // Fallback: cooperative LDS fill (float4 chunks)
    for (unsigned i = threadIdx.x * 4; i < D_IN * NC; i += blockDim.x * 4)
        *(float4*)&wlds[i] = *(const float4*)&Wm[i];
    __syncthreads();
#endif

    const int wave = threadIdx.x >> 5;
    const int lane = threadIdx.x & 31;
    const int tile = blockIdx.x * 8 + wave;
    if (tile >= rowTiles) return;                  // wave-uniform: EXEC all-ones inside WMMA

    const int n16 = lane & 15;                     // M for A-frag, N for B/C-frags
    const int kq  = (lane >> 4) << 1;              // K-pair select per ISA A/B layout

    const float* xrow = X + (size_t)(tile * 16 + n16) * D_IN;

    v8f acc[CT] = {};
    for (int k = 0; k < D_IN; k += 4) {
        v2f a = *(const v2f*)(xrow + k + kq);      // 8B-aligned global_load_b64
#pragma unroll
        for (int t = 0; t < CT; ++t) {
            v2f b;
            b.x = wlds[(k + kq)     * NC + t * 16 + n16];
            b.y = wlds[(k + kq + 1) * NC + t * 16 + n16];
            acc[t] = __builtin_amdgcn_wmma_f32_16x16x4_f32(
                false, a, false, b, (short)0, acc[t], false, false);
        }
    }

    const int mh = (lane >> 4) << 3;               // C/D: M = r + 8*(lane>=16)
#pragma unroll
    for (int t = 0; t < CT; ++t)
#pragma unroll
        for (int r = 0; r < 8; ++r)
            Out[(size_t)(tile * 16 + mh + r) * NC + t * 16 + n16] = acc[t][r];
}

// ---------------- edge scatter (message passing) ----------------
// One wave per edge (plus N virtual self-loop edges).

__global__ void k_scatter128(const float* __restrict__ H, const int* __restrict__ src,
                             const int* __restrict__ dst, const float* __restrict__ ew,
                             const float* __restrict__ dinv, float* __restrict__ Out,
                             int E, int N) {
    long long gw = ((long long)blockIdx.x * blockDim.x + threadIdx.x) >> 5;
    int lane = threadIdx.x & 31;
    if (gw >= (long long)E + N) return;
    int s, d; float w;
    if (gw < E) { s = src[gw]; d = dst[gw]; w = ew[gw]; }
    else        { s = d = (int)(gw - E);    w = 1.0f;   }
    float norm = dinv[s] * w * dinv[d];
    float4 v = ((const float4*)(H + (size_t)s * D_HID))[lane];
    float* orow = Out + (size_t)d * D_HID + lane * 4;
    unsafeAtomicAdd(orow + 0, v.x * norm);
    unsafeAtomicAdd(orow + 1, v.y * norm);
    unsafeAtomicAdd(orow + 2, v.z * norm);
    unsafeAtomicAdd(orow + 3, v.w * norm);
}

__global__ void k_scatter40(const float* __restrict__ H, const int* __restrict__ src,
                            const int* __restrict__ dst, const float* __restrict__ ew,
                            const float* __restrict__ dinv, float* __restrict__ Out,
                            int E, int N) {
    long long gw = ((long long)blockIdx.x * blockDim.x + threadIdx.x) >> 5;
    int lane = threadIdx.x & 31;
    if (gw >= (long long)E + N) return;
    int s, d; float w;
    if (gw < E) { s = src[gw]; d = dst[gw]; w = ew[gw]; }
    else        { s = d = (int)(gw - E);    w = 1.0f;   }
    float norm = dinv[s] * w * dinv[d];
    const float* hrow = H + (size_t)s * D_OUT_PAD;
    float* orow = Out + (size_t)d * D_OUT;
    for (int f = lane; f < D_OUT; f += 32)
        unsafeAtomicAdd(orow + f, hrow[f] * norm);
}

// ---------------- pointwise ----------------

__global__ void k_bias_relu128(float* __restrict__ A, const float* __restrict__ b, long long n) {
    long long i = (long long)blockIdx.x * blockDim.x + threadIdx.x;
    if (i < n) {
        float v = A[i] + b[i & (D_HID - 1)];
        A[i] = v > 0.0f ? v : 0.0f;
    }
}

__global__ void k_pad_w2(const float* __restrict__ W2, float* __restrict__ Wp) {
    int i = blockIdx.x * blockDim.x + threadIdx.x;          // over 128*48
    if (i < D_HID * D_OUT_PAD) {
        int k = i / D_OUT_PAD, c = i % D_OUT_PAD;
        Wp[i] = (c < D_OUT) ? W2[k * D_OUT + c] : 0.0f;
    }
}

// one wave per node: bias + log_softmax over 40 features
__global__ void k_bias_logsoftmax(const float* __restrict__ A, const float* __restrict__ b,
                                  float* __restrict__ Out, int N) {
    long long node = ((long long)blockIdx.x * blockDim.x + threadIdx.x) >> 5;
    int lane = threadIdx.x & 31;
    if (node >= N) return;
    const float* row = A + (size_t)node * D_OUT;
    float x0 = row[lane] + b[lane];
    float x1 = (lane < D_OUT - 32) ? row[lane + 32] + b[lane + 32] : -INFINITY;
    float m = fmaxf(x0, x1);
    for (int off = 16; off > 0; off >>= 1) m = fmaxf(m, __shfl_xor(m, off, 32));
    float e = expf(x0 - m) + ((lane < D_OUT - 32) ? expf(x1 - m) : 0.0f);
    for (int off = 16; off > 0; off >>= 1) e += __shfl_xor(e, off, 32);
    float lse = m + logf(e);
    float* orow = Out + (size_t)node * D_OUT;
    orow[lane] = x0 - lse;
    if (lane < D_OUT - 32) orow[lane + 32] = x1 - lse;
}

// ---------------- launch ----------------

extern "C" void kernel_launch(void* const* d_in, const int* in_sizes, int n_in,
                              void* d_out, int out_size, void* d_ws, size_t ws_size,
                              hipStream_t stream) {
    const float* x  = (const float*)d_in[0];
    const int*   ei = (const int*)d_in[1];
    const float* ew = (const float*)d_in[2];
    const float* W1 = (const float*)d_in[3];
    const float* b1 = (const float*)d_in[4];
    const float* W2 = (const float*)d_in[5];
    const float* b2 = (const float*)d_in[6];
    float* out = (float*)d_out;

    const int N = in_sizes[0] / D_IN;
    const int E = in_sizes[1] / 2;
    const int* src = ei;
    const int* dst = ei + E;

    char* w = (char*)d_ws;
    auto alloc = [&](size_t bytes) {
        char* p = w; w += (bytes + 255) & ~(size_t)255; return (float*)p;
    };
    float* dinv = alloc((size_t)N * 4);                       // deg -> dinv in place
    float* h1   = alloc((size_t)N * D_HID * 4);
    float* a1   = alloc((size_t)N * D_HID * 4);
    float* wpad = alloc((size_t)D_HID * D_OUT_PAD * 4);
    float* h2   = alloc((size_t)N * D_OUT_PAD * 4);
    float* a2   = alloc((size_t)N * D_OUT * 4);
    (void)ws_size; (void)n_in; (void)out_size;

    const int rowTiles  = (N + 15) / 16;
    const long long tot = (long long)E + N;                   // edges + self-loops
    const int scatBlks  = (int)((tot * 32 + 255) / 256);
    dim3 B(256);

    // normalization: deg = 1 + segment_sum(w, dst); dinv = rsqrt(deg)
    k_init_deg<<<(N + 255) / 256, B, 0, stream>>>(dinv, N);
    k_deg_scatter<<<(E + 255) / 256, B, 0, stream>>>(dinv, dst, ew, E);
    k_rsqrt_inplace<<<(N + 255) / 256, B, 0, stream>>>(dinv, N);

    // layer 1: h1 = x @ W1 ; a1 = scatter(norm * h1[src]) ; relu(a1 + b1)
    k_gemm_wmma<D_HID / 16><<<(rowTiles + 7) / 8, B, 0, stream>>>(x, W1, h1, rowTiles);
    k_zero<<<(int)(((long long)N * D_HID + 255) / 256), B, 0, stream>>>(a1, (long long)N * D_HID);
    k_scatter128<<<scatBlks, B, 0, stream>>>(h1, src, dst, ew, dinv, a1, E, N);
    k_bias_relu128<<<(int)(((long long)N * D_HID + 255) / 256), B, 0, stream>>>(a1, b1, (long long)N * D_HID);

    // layer 2: h2 = a1 @ pad(W2) ; a2 = scatter ; log_softmax(a2 + b2)
    k_pad_w2<<<(D_HID * D_OUT_PAD + 255) / 256, B, 0, stream>>>(W2, wpad);
    k_gemm_wmma<D_OUT_PAD / 16><<<(rowTiles + 7) / 8, B, 0, stream>>>(a1, wpad, h2, rowTiles);
    k_zero<<<(int)(((long long)N * D_OUT + 255) / 256), B, 0, stream>>>(a2, (long long)N * D_OUT);
    k_scatter40<<<scatBlks, B, 0, stream>>>(h2, src, dst, ew, dinv, a2, E, N);
    k_bias_logsoftmax<<<(int)(((long long)N * 32 + 255) / 256), B, 0, stream>>>(a2, b2, out, N);
}